// TensorGraphConvolution_48988396978752
// MI455X (gfx1250) — compile-verified
//
#include <hip/hip_runtime.h>

// Problem constants (fixed by the reference).
#define T_  4
#define N_  50000
#define F_  64
#define E_  800000
#define O_  4
#define NF_ (N_ * F_)          // 3,200,000 elements per channel
#define NTILES_ (T_ * (N_ / 16))  // 12,500 WMMA row-tiles (N divisible by 16)

typedef float v2f __attribute__((ext_vector_type(2)));
typedef float v8f __attribute__((ext_vector_type(8)));

// ---------------------------------------------------------------------------
// Kernel 1: Xt[i,n,f] = sum_j transm[i,j] * x[j,n,f]   (4x4 channel mix)
// Pure streaming, float4-vectorized. 1 thread = 4 consecutive f.
// ---------------------------------------------------------------------------
__global__ __launch_bounds__(256) void k_chanmix(const float* __restrict__ x,
                                                 const float* __restrict__ transm,
                                                 float* __restrict__ Xt) {
    int tid = blockIdx.x * 256 + threadIdx.x;
    if (tid >= NF_ / 4) return;

    float m[T_][T_];
#pragma unroll
    for (int i = 0; i < T_; ++i)
#pragma unroll
        for (int j = 0; j < T_; ++j)
            m[i][j] = transm[i * T_ + j];

    const size_t nf = (size_t)tid * 4;
    float4 xj[T_];
#pragma unroll
    for (int j = 0; j < T_; ++j)
        xj[j] = *reinterpret_cast<const float4*>(x + (size_t)j * NF_ + nf);

#pragma unroll
    for (int i = 0; i < T_; ++i) {
        float4 o;
        o.x = m[i][0] * xj[0].x + m[i][1] * xj[1].x + m[i][2] * xj[2].x + m[i][3] * xj[3].x;
        o.y = m[i][0] * xj[0].y + m[i][1] * xj[1].y + m[i][2] * xj[2].y + m[i][3] * xj[3].y;
        o.z = m[i][0] * xj[0].z + m[i][1] * xj[1].z + m[i][2] * xj[2].z + m[i][3] * xj[3].z;
        o.w = m[i][0] * xj[0].w + m[i][1] * xj[1].w + m[i][2] * xj[2].w + m[i][3] * xj[3].w;
        *reinterpret_cast<float4*>(Xt + (size_t)i * NF_ + nf) = o;
    }
}

// ---------------------------------------------------------------------------
// Kernel 2: AtXt[i,row_e,f] += transm[i,chan_e]*val_e * Xt[i,col_e,f]
// One wave32 per edge; lane l handles f = 2l, 2l+1 (float2). Xt and AtXt are
// both L2-resident (51.2 MB each vs 192 MB L2), atomics land in L2.
// ---------------------------------------------------------------------------
__global__ __launch_bounds__(256) void k_scatter(const float* __restrict__ Xt,
                                                 const float* __restrict__ transm,
                                                 const float* __restrict__ edge_val,
                                                 const int* __restrict__ echan,
                                                 const int* __restrict__ erow,
                                                 const int* __restrict__ ecol,
                                                 float* __restrict__ AtXt) {
    const int wave = threadIdx.x >> 5;
    const int lane = threadIdx.x & 31;
    const long long e = (long long)blockIdx.x * 8 + wave;
    if (e >= E_) return;

    const int chan = echan[e];
    const int row  = erow[e];
    const int col  = ecol[e];
    const float val = edge_val[e];
    const int f0 = lane * 2;

#pragma unroll
    for (int i = 0; i < T_; ++i) {
        const float coef = transm[i * T_ + chan] * val;
        const float2 xv = *reinterpret_cast<const float2*>(
            Xt + (size_t)(i * N_ + col) * F_ + f0);
        float* dst = AtXt + (size_t)(i * N_ + row) * F_ + f0;
        unsafeAtomicAdd(dst,     coef * xv.x);  // global_atomic_add_f32
        unsafeAtomicAdd(dst + 1, coef * xv.y);
    }
}

// ---------------------------------------------------------------------------
// Kernel 3: out[i,n,o] = sum_f AtXt[i,n,f] * W[i,f,o] + bias[i,n,o]
// Per-channel [N,64]x[64,4] GEMM via chained V_WMMA_F32_16X16X4_F32.
// One wave = one (channel, 16-row) tile; K=64 contracted as 16 k=4 steps.
// A fragment (16x4 f32): lane (m=l&15, half=l>>4) holds float2 at f=4k+2*half.
// B fragment (4x16 f32): lane holds W[i, 4k+2*half + {0,1}, ncol] (ncol=l&15),
// columns >= O_ zeroed BRANCHLESSLY (clamped address + mask multiply) so the
// inner loop is a clean load_b64 + v_wmma chain with full EXEC throughout.
// All 16 B fragments are hoisted out of the k-loop (weight is 1 KB/channel,
// L1-resident; ~40 VGPRs of live state total).
// ---------------------------------------------------------------------------
__global__ __launch_bounds__(256) void k_wmma_out(const float* __restrict__ AtXt,
                                                  const float* __restrict__ W,
                                                  const float* __restrict__ bias,
                                                  float* __restrict__ out) {
    const int wave = threadIdx.x >> 5;
    const int lane = threadIdx.x & 31;
    const int tile = blockIdx.x * 8 + wave;
    if (tile >= NTILES_) return;           // wave-uniform: EXEC stays all-ones

    const int i  = tile / (N_ / 16);
    const int n0 = (tile % (N_ / 16)) * 16;
    const int m    = lane & 15;            // A: row within tile;  B/D: column
    const int half = lane >> 4;

    const int   mc    = m & (O_ - 1);                // clamped, always in-bounds
    const float wmask = (m < O_) ? 1.0f : 0.0f;      // v_cndmask, no branch

    // Hoist all 16 B fragments (branchless, non-divergent loads).
    const float* wbase = W + (size_t)i * F_ * O_ + mc;
    v2f bfrag[16];
#pragma unroll
    for (int k = 0; k < 16; ++k) {
        const int kk0 = 4 * k + 2 * half;
        bfrag[k].x = wmask * wbase[(size_t)kk0 * O_];
        bfrag[k].y = wmask * wbase[(size_t)(kk0 + 1) * O_];
    }

    const float* arow = AtXt + (size_t)(i * N_ + n0 + m) * F_;
    v8f acc = {};

#pragma unroll
    for (int k = 0; k < 16; ++k) {
        const float2 av = *reinterpret_cast<const float2*>(arow + 4 * k + 2 * half);
        v2f a; a.x = av.x; a.y = av.y;
        acc = __builtin_amdgcn_wmma_f32_16x16x4_f32(
            /*neg_a=*/false, a, /*neg_b=*/false, bfrag[k],
            /*c_mod=*/(short)0, acc, /*reuse_a=*/false, /*reuse_b=*/false);
    }

    if (m < O_) {
#pragma unroll
        for (int r = 0; r < 8; ++r) {
            const int n = n0 + r + 8 * half;
            const size_t o = (size_t)(i * N_ + n) * O_ + m;
            out[o] = acc[r] + bias[o];
        }
    }
}

// ---------------------------------------------------------------------------
extern "C" void kernel_launch(void* const* d_in, const int* in_sizes, int n_in,
                              void* d_out, int out_size, void* d_ws, size_t ws_size,
                              hipStream_t stream) {
    const float* x        = (const float*)d_in[0];   // [T,N,F]
    const float* edge_val = (const float*)d_in[1];   // [E]
    const float* transm   = (const float*)d_in[2];   // [T,T]
    const float* weight   = (const float*)d_in[3];   // [T,F,O]
    const float* bias     = (const float*)d_in[4];   // [T,N,O]
    const int*   echan    = (const int*)d_in[5];     // [E]
    const int*   erow     = (const int*)d_in[6];     // [E]
    const int*   ecol     = (const int*)d_in[7];     // [E]
    float* out = (float*)d_out;                      // [T,N,O]

    float* Xt   = (float*)d_ws;                      // 51.2 MB
    float* AtXt = Xt + (size_t)T_ * NF_;             // 51.2 MB

    hipMemsetAsync(AtXt, 0, (size_t)T_ * NF_ * sizeof(float), stream);

    k_chanmix<<<(NF_ / 4 + 255) / 256, 256, 0, stream>>>(x, transm, Xt);

    k_scatter<<<E_ / 8, 256, 0, stream>>>(Xt, transm, edge_val,
                                          echan, erow, ecol, AtXt);

    k_wmma_out<<<(NTILES_ + 7) / 8, 256, 0, stream>>>(AtXt, weight, bias, out);
}